// GNN_61280593379641
// MI455X (gfx1250) — compile-verified
//
#include <hip/hip_runtime.h>
#include <hip/hip_bf16.h>

typedef __attribute__((ext_vector_type(2))) float v2f;
typedef __attribute__((ext_vector_type(8))) float v8f;

#define HIDC 64
#define NCLS 10
#define MAXK 128
#define QSTR 80   // LDS row stride in float2 (64 data + 16 pad) -> half-waves on disjoint banks

// ---------------- GEMM: ht[N x 64] = (X[N x K] @ W[K x 64]) * dinv[row] ----
// One wave computes a 16-row x 64-col strip with V_WMMA_F32_16X16X4_F32.
// W staged in LDS as interleaved (k,k+1) float2 pairs: each B fragment is one
// aligned ds_load_b64. Epilogue scales rows by dinv and writes to BOTH hA (ht,
// gather source) and hB (aggregation accumulator init = self-loop term).
__global__ void __launch_bounds__(256) gcn_gemm_wmma(
    const float* __restrict__ X, const float* __restrict__ W,
    const float* __restrict__ dinv,
    float* __restrict__ hA, float* __restrict__ hB, int N, int K)
{
  __shared__ float Wlds[(MAXK / 2) * QSTR * 2];

  // stage W: Wlds pair[m][n] = (W[2m][n], W[2m+1][n]), row stride QSTR float2
  for (int i = threadIdx.x; i < K * HIDC; i += 256) {
    int kk = i >> 6, n = i & (HIDC - 1);
    Wlds[((kk >> 1) * QSTR + n) * 2 + (kk & 1)] = W[i];
  }
  __syncthreads();

  const int lane = threadIdx.x & 31;
  const int wave = blockIdx.x * 8 + (threadIdx.x >> 5);
  const int rowBase = wave * 16;
  if (rowBase >= N) return;
  const int half = lane >> 4;   // 0: lanes 0-15, 1: lanes 16-31
  const int l16  = lane & 15;

  v8f zero = {0.f,0.f,0.f,0.f,0.f,0.f,0.f,0.f};
  v8f acc[4];
  #pragma unroll
  for (int t = 0; t < 4; ++t) acc[t] = zero;

  const v2f* __restrict__ Wp = (const v2f*)Wlds;
  const float* __restrict__ xrow = X + (size_t)(rowBase + l16) * K;
  for (int k = 0; k < K; k += 4) {
    const int ka = k + 2 * half;          // A cols (k) ka, ka+1 for this half-wave
    const int m  = (k >> 1) + half;       // B pair index
    v2f a;
    a.x = xrow[ka];
    a.y = xrow[ka + 1];
    #pragma unroll
    for (int t = 0; t < 4; ++t) {
      v2f b = Wp[(size_t)m * QSTR + t * 16 + l16];   // ds_load_b64
      acc[t] = __builtin_amdgcn_wmma_f32_16x16x4_f32(
          false, a, false, b, (short)0, acc[t], false, false);
    }
  }

  #pragma unroll
  for (int r = 0; r < 8; ++r) {
    const int row = rowBase + r + 8 * half;
    const float dr = dinv[row];
    #pragma unroll
    for (int t = 0; t < 4; ++t) {
      const float v = acc[t][r] * dr;
      const size_t idx = (size_t)row * HIDC + t * 16 + l16;
      hA[idx] = v;   // ht: gather source for edge scatter
      hB[idx] = v;   // self-loop term = accumulator init
    }
  }
}

// ---------------- degree / norm / edge pack ----------------
__global__ void deg_init(float* deg, int N) {
  int i = blockIdx.x * blockDim.x + threadIdx.x;
  if (i < N) deg[i] = 1.0f;  // self loop
}
__global__ void deg_accum(const long long* __restrict__ ei, float* deg, int E) {
  int e = blockIdx.x * blockDim.x + threadIdx.x;
  if (e < E) atomicAdd(&deg[(int)ei[(size_t)E + e]], 1.0f);
}
__global__ void deg_rsqrt(float* deg, int N) {
  int i = blockIdx.x * blockDim.x + threadIdx.x;
  if (i < N) { float d = deg[i]; deg[i] = d > 0.f ? rsqrtf(d) : 0.f; }
}
__global__ void edge_pack(const long long* __restrict__ ei, int2* __restrict__ sd,
                          int E) {
  int e = blockIdx.x * blockDim.x + threadIdx.x;
  if (e < E) sd[e] = make_int2((int)ei[e], (int)ei[(size_t)E + e]);
}

// ---------------- edge scatter: out[d] += ht[s] (pure gather + atomicAdd) ---
// one wave per edge; each lane handles 2 features (float2 gather + 2 f32 atomics)
__global__ void __launch_bounds__(256) edge_scatter(
    const float* __restrict__ ht, const int2* __restrict__ sd,
    float* __restrict__ out, int E)
{
  int wave = blockIdx.x * (blockDim.x >> 5) + (threadIdx.x >> 5);
  if (wave >= E) return;
  int lane = threadIdx.x & 31;
  int2 e = sd[wave];                                  // single b64
  const float2* __restrict__ hs = (const float2*)(ht + (size_t)e.x * HIDC);
  float2 v = hs[lane];
  float* outd = out + (size_t)e.y * HIDC + 2 * lane;
  atomicAdd(outd,     v.x);
  atomicAdd(outd + 1, v.y);
}

// ---------------- h = act(dinv[row] * h + b) (float4 vectorized) -----------
__global__ void bias_act(float* __restrict__ h, const float* __restrict__ dinv,
                         const float* __restrict__ b, int N, int relu) {
  int i = blockIdx.x * blockDim.x + threadIdx.x;   // over N*16 float4s
  if (i < N * (HIDC / 4)) {
    int n  = i >> 4;
    int f4 = i & 15;
    float d = dinv[n];
    float4 v  = ((const float4*)h)[i];
    float4 bb = ((const float4*)b)[f4];
    v.x = v.x * d + bb.x;  v.y = v.y * d + bb.y;
    v.z = v.z * d + bb.z;  v.w = v.w * d + bb.w;
    if (relu) {
      v.x = fmaxf(v.x, 0.f); v.y = fmaxf(v.y, 0.f);
      v.z = fmaxf(v.z, 0.f); v.w = fmaxf(v.w, 0.f);
    }
    ((float4*)h)[i] = v;
  }
}

// ---------------- pooling + FC ----------------
__global__ void pool_zero(float* gsum, float* gcnt, int G) {
  int i = blockIdx.x * blockDim.x + threadIdx.x;
  if (i < G * HIDC) gsum[i] = 0.f;
  if (i < G) gcnt[i] = 0.f;
}
__global__ void pool_accum(const float* __restrict__ h,
                           const long long* __restrict__ batch,
                           float* gsum, float* gcnt, int N) {
  int i = blockIdx.x * blockDim.x + threadIdx.x;
  if (i < N * HIDC) {
    int n = i >> 6, f = i & (HIDC - 1);
    int g = (int)batch[n];
    atomicAdd(&gsum[g * HIDC + f], h[i]);
    if (f == 0) atomicAdd(&gcnt[g], 1.0f);
  }
}
__global__ void fc_kernel(const float* __restrict__ gsum,
                          const float* __restrict__ gcnt,
                          const float* __restrict__ Wfc,
                          const float* __restrict__ bfc,
                          float* __restrict__ out, int G) {
  int i = blockIdx.x * blockDim.x + threadIdx.x;
  if (i < G * NCLS) {
    int g = i / NCLS, c = i % NCLS;
    float inv = 1.0f / fmaxf(gcnt[g], 1.0f);
    float a = 0.f;
    #pragma unroll
    for (int f = 0; f < HIDC; ++f)
      a += (gsum[g * HIDC + f] * inv) * Wfc[f * NCLS + c];
    out[i] = a + bfc[c];
  }
}

// ---------------- driver ----------------
extern "C" void kernel_launch(void* const* d_in, const int* in_sizes, int n_in,
                              void* d_out, int out_size, void* d_ws, size_t ws_size,
                              hipStream_t stream) {
  const float*     x     = (const float*)d_in[0];
  const long long* ei    = (const long long*)d_in[1];   // [2, E] int64
  const long long* batch = (const long long*)d_in[2];   // [N]    int64
  const float* W1  = (const float*)d_in[3];
  const float* b1  = (const float*)d_in[4];
  const float* W2  = (const float*)d_in[5];
  const float* b2  = (const float*)d_in[6];
  const float* W3  = (const float*)d_in[7];
  const float* b3  = (const float*)d_in[8];
  const float* Wfc = (const float*)d_in[9];
  const float* bfc = (const float*)d_in[10];

  const int N  = in_sizes[2];            // batch has N elements
  const int E  = in_sizes[1] / 2;
  const int G  = out_size / NCLS;
  const int IN = in_sizes[0] / N;        // 128

  // workspace layout (everything read is written first each call)
  auto al = [](size_t v) { return (v + 255) & ~(size_t)255; };
  char* p = (char*)d_ws;
  float* dinv = (float*)p; p += al((size_t)N * 4);
  float* hA   = (float*)p; p += al((size_t)N * HIDC * 4);   // ht (gather source)
  float* hB   = (float*)p; p += al((size_t)N * HIDC * 4);   // aggregation buffer
  int2*  sd   = (int2*) p; p += al((size_t)E * 8);          // packed (src,dst)
  float* gsum = (float*)p; p += al((size_t)G * HIDC * 4);
  float* gcnt = (float*)p;

  const int gemmBlocks = ((N / 16) + 7) / 8;        // 8 waves/block, 16 rows/wave
  const int edgeBlocks = (E + 7) / 8;               // 8 waves/block, 1 edge/wave
  const int eltBlocks  = (N * (HIDC / 4) + 255) / 256;

  // one-time prep: degrees -> dinv, pack edges to int2
  deg_init <<<(N + 255) / 256, 256, 0, stream>>>(dinv, N);
  deg_accum<<<(E + 255) / 256, 256, 0, stream>>>(ei, dinv, E);
  deg_rsqrt<<<(N + 255) / 256, 256, 0, stream>>>(dinv, N);
  edge_pack<<<(E + 255) / 256, 256, 0, stream>>>(ei, sd, E);

  // ---- layer 1 ----
  gcn_gemm_wmma<<<gemmBlocks, 256, 0, stream>>>(x, W1, dinv, hA, hB, N, IN);
  edge_scatter <<<edgeBlocks, 256, 0, stream>>>(hA, sd, hB, E);
  bias_act     <<<eltBlocks, 256, 0, stream>>>(hB, dinv, b1, N, 1);

  // ---- layer 2 (reads hB, overwrites hB in epilogue: wave-local rows only) ----
  gcn_gemm_wmma<<<gemmBlocks, 256, 0, stream>>>(hB, W2, dinv, hA, hB, N, HIDC);
  edge_scatter <<<edgeBlocks, 256, 0, stream>>>(hA, sd, hB, E);
  bias_act     <<<eltBlocks, 256, 0, stream>>>(hB, dinv, b2, N, 1);

  // ---- layer 3 (no ReLU) ----
  gcn_gemm_wmma<<<gemmBlocks, 256, 0, stream>>>(hB, W3, dinv, hA, hB, N, HIDC);
  edge_scatter <<<edgeBlocks, 256, 0, stream>>>(hA, sd, hB, E);
  bias_act     <<<eltBlocks, 256, 0, stream>>>(hB, dinv, b3, N, 0);

  // ---- pool + FC ----
  pool_zero <<<(G * HIDC + 255) / 256, 256, 0, stream>>>(gsum, gcnt, G);
  pool_accum<<<(N * HIDC + 255) / 256, 256, 0, stream>>>(hB, batch, gsum, gcnt, N);
  fc_kernel <<<(G * NCLS + 255) / 256, 256, 0, stream>>>(gsum, gcnt, Wfc, bfc,
                                                         (float*)d_out, G);
}